// Lattice_11836929867945
// MI455X (gfx1250) — compile-verified
//
#include <hip/hip_runtime.h>

// Problem constants from the reference
#define PROT_N 64          // protein length
#define LAT_L  127         // lattice side = 2N-1
#define SHIFT  (PROT_N-1)  // index shift = 63

// ---------------------------------------------------------------------------
// Kernel 1: zero the 262 MB lattice by streaming 128-bit async stores of a
// zeroed LDS staging buffer to global memory (CDNA5 async LDS->global path,
// tracked with ASYNCcnt). Non-temporal hint: output (262MB) > L2 (192MB).
// ---------------------------------------------------------------------------
__global__ void lattice_zero_async(float* __restrict__ out, long long n4, int rem) {
  __shared__ __align__(16) float zbuf[1024];           // 4 KB of zeros
  const int tid = threadIdx.x;
  float4* z4 = reinterpret_cast<float4*>(zbuf);
  z4[tid] = make_float4(0.f, 0.f, 0.f, 0.f);
  __syncthreads();

  // Flat address of an LDS location truncated to 32 bits == DS byte offset
  // (aperture bits live in the high dword).
  unsigned lds_addr = (unsigned)(unsigned long long)(uintptr_t)&z4[tid];

  const long long stride = (long long)gridDim.x * (long long)blockDim.x;
  for (long long i = (long long)blockIdx.x * blockDim.x + tid; i < n4; i += stride) {
    unsigned long long gaddr =
        (unsigned long long)(uintptr_t)out + ((unsigned long long)i << 4);
    asm volatile("global_store_async_from_lds_b128 %0, %1, off th:TH_STORE_NT"
                 :: "v"(gaddr), "v"(lds_addr)
                 : "memory");
  }
  // Drain outstanding async stores before wave end.
  asm volatile("s_wait_asynccnt 0x0" ::: "memory");

  // Scalar tail (out_size % 4); zero for this problem but kept for safety.
  if (blockIdx.x == 0 && tid < rem) {
    out[(n4 << 2) + tid] = 0.0f;
  }
}

// ---------------------------------------------------------------------------
// Kernel 2: masked scatter-add of 2048 values. Duplicate sites are possible
// within a batch -> hardware float atomics (global_atomic_add_f32).
// ---------------------------------------------------------------------------
__global__ void lattice_scatter(const float* __restrict__ acids,
                                const int* __restrict__ idx,
                                const unsigned char* __restrict__ mask,
                                float* __restrict__ out, int bn) {
  int t = blockIdx.x * blockDim.x + threadIdx.x;
  if (t >= bn) return;
  if (!mask[t]) return;            // masked-off: scatter 0 == no-op, skip

  float v = acids[t];
  int x = idx[3 * t + 0] + SHIFT;
  int y = idx[3 * t + 1] + SHIFT;
  int z = idx[3 * t + 2] + SHIFT;
  int b = t >> 6;                  // N = 64

  long long off = (((long long)b * LAT_L + x) * LAT_L + y) * LAT_L + z;
  unsafeAtomicAdd(out + off, v);   // hardware fp32 atomic add
}

// ---------------------------------------------------------------------------
extern "C" void kernel_launch(void* const* d_in, const int* in_sizes, int n_in,
                              void* d_out, int out_size, void* d_ws, size_t ws_size,
                              hipStream_t stream) {
  (void)n_in; (void)d_ws; (void)ws_size;

  const float*         acids = (const float*)d_in[0];         // (B,N) f32
  const int*           idx   = (const int*)d_in[1];           // (B,N,3) i32
  const unsigned char* mask  = (const unsigned char*)d_in[2]; // (B,N) bool (1B)
  float* out = (float*)d_out;                                 // (B,L,L,L) f32

  long long total = (long long)out_size;   // 65,548,256
  long long n4    = total >> 2;            // float4 chunks
  int       rem   = (int)(total & 3);      // 0 here

  // Zero pass: 4096 blocks x 256 threads (8 wave32 per block) grid-striding
  // 16,387,064 float4 stores -> saturates HBM write bandwidth.
  lattice_zero_async<<<4096, 256, 0, stream>>>(out, n4, rem);

  // Scatter pass: one lane per (b, n).
  int bn = in_sizes[0];                    // B*N = 2048
  lattice_scatter<<<(bn + 255) / 256, 256, 0, stream>>>(acids, idx, mask, out, bn);
}